// KLDivFullGaussian_13486197310107
// MI455X (gfx1250) — compile-verified
//
#include <hip/hip_runtime.h>

#define LSTRIDE 41   // padded LDS row stride (floats): odd, and 2*41 % 64 = 18 -> conflict-free WMMA operand gathers
#define EPSV 1e-3f
#define WPB 4        // waves per block

typedef float v2f __attribute__((ext_vector_type(2)));
typedef float v8f __attribute__((ext_vector_type(8)));

__device__ __forceinline__ float lane_bcast(float v, int lane) {
  return __builtin_bit_cast(float, __builtin_amdgcn_readlane(__builtin_bit_cast(int, v), lane));
}

// Right-looking Cholesky of a 32x32 SPD matrix stored row-per-lane
// (lane i holds row i; register a[j] = column j across lanes).
// eps is folded into each pivot (exactly equivalent to factoring A + eps*I,
// since diagonal entries only enter the factorization through pivots).
// Returns logdet(A + eps*I) (uniform across lanes). On exit a[] holds L in
// the lower triangle (upper triangle garbage, never read).
__device__ __forceinline__ float chol32(float (&a)[32], int lane) {
  float ld = 0.0f;
#pragma unroll
  for (int k = 0; k < 32; ++k) {
    float akk = lane_bcast(a[k], k) + EPSV;    // pivot (+eps)
    ld += __logf(akk);                         // 2*log(Lkk) = log(akk)
    float is = __builtin_amdgcn_rsqf(akk);     // NaN if pivot <= 0 -> propagates (torch.logdet semantics)
    float s  = akk * is;                       // Lkk = sqrt(akk)
    float sc = a[k] * is;
    a[k] = (lane == k) ? s : sc;               // finalize column k of L
#pragma unroll
    for (int j = k + 1; j < 32; ++j) {
      float ljk = lane_bcast(a[k], j);         // L[j][k]
      a[j] = __builtin_fmaf(-ljk, a[k], a[j]); // trailing rank-1 update
    }
  }
  return ld;
}

__global__ void __launch_bounds__(32 * WPB)
kl_full_gaussian_kernel(const float* __restrict__ p_mu,
                        const float* __restrict__ p_cov,
                        const float* __restrict__ q_mu,
                        const float* __restrict__ q_cov,
                        float* __restrict__ out, int nbatch) {
  __shared__ float lds[WPB][2][32 * LSTRIDE];  // ~41 KB static
  const int lane = threadIdx.x & 31;
  const int wid  = threadIdx.x >> 5;
  const int b    = blockIdx.x * WPB + wid;     // one wave32 per batch element
  if (b >= nbatch) return;                     // uniform per wave: EXEC stays all-1s for WMMA

  // ---- load covariance rows (row-per-lane): 8x global_load_b128 each ----
  float aq[32], ap[32];
  const size_t mb = (size_t)b * 1024 + (size_t)lane * 32;
  const float4* qv = (const float4*)(q_cov + mb);
  const float4* pv = (const float4*)(p_cov + mb);
#pragma unroll
  for (int j = 0; j < 8; ++j) {
    float4 t = qv[j];
    aq[4*j+0] = t.x; aq[4*j+1] = t.y; aq[4*j+2] = t.z; aq[4*j+3] = t.w;
  }
#pragma unroll
  for (int j = 0; j < 8; ++j) {
    float4 t = pv[j];
    ap[4*j+0] = t.x; ap[4*j+1] = t.y; ap[4*j+2] = t.z; ap[4*j+3] = t.w;
  }
  float dv = q_mu[(size_t)b * 32 + lane] - p_mu[(size_t)b * 32 + lane];

  // ---- Cholesky both matrices; logdets come for free ----
  float ldq = chol32(aq, lane);
  float ldp = chol32(ap, lane);

  // ---- Ti = Lq^{-1}: column-per-lane forward substitution (lane c = column c) ----
  float ti[32];
#pragma unroll
  for (int k = 0; k < 32; ++k) {
    float t = (lane == k) ? 1.0f : 0.0f;
#pragma unroll
    for (int j = 0; j < k; ++j) {
      float lkj = lane_bcast(aq[j], k);        // Lq[k][j]
      t = __builtin_fmaf(-lkj, ti[j], t);
    }
    float lkk = lane_bcast(aq[k], k);          // = sqrt(akk+eps)
    ti[k] = t * __builtin_amdgcn_rcpf(lkk);    // ti[k]=0 for rows above the diagonal by induction
  }

  // ---- stage Ti and Lp (upper triangle zeroed) in LDS, row-major, stride 41 ----
  float* Ti = &lds[wid][0][0];
  float* Lp = &lds[wid][1][0];
#pragma unroll
  for (int k = 0; k < 32; ++k)
    Ti[k * LSTRIDE + lane] = ti[k];                        // Ti[k][lane]
#pragma unroll
  for (int j = 0; j < 32; ++j)
    Lp[lane * LSTRIDE + j] = (lane >= j) ? ap[j] : 0.0f;   // Lp[lane][j]

  // ---- X = Ti * Lp via V_WMMA_F32_16X16X4_F32; trace = ||X||_F^2 ----
  // Lower-triangular block structure: X00 = Ti00*Lp00; X10 = Ti10*Lp00 + Ti11*Lp10; X11 = Ti11*Lp11.
  const int m = lane & 15;
  const int h = lane >> 4;
  auto mm16 = [&](v8f acc, int I, int J, int kb) -> v8f {
#pragma unroll
    for (int kk = 0; kk < 4; ++kk) {
      const int ka = 16 * kb + 4 * kk + 2 * h;   // A: VGPR0 holds K=0/K=2, VGPR1 holds K=1/K=3 per half
      v2f A, Bv;
      A[0]  = Ti[(16 * I + m) * LSTRIDE + ka + 0];
      A[1]  = Ti[(16 * I + m) * LSTRIDE + ka + 1];
      Bv[0] = Lp[(ka + 0) * LSTRIDE + 16 * J + m];
      Bv[1] = Lp[(ka + 1) * LSTRIDE + 16 * J + m];
      acc = __builtin_amdgcn_wmma_f32_16x16x4_f32(false, A, false, Bv,
                                                  (short)0, acc, false, false);
    }
    return acc;
  };
  v8f acc00 = {0,0,0,0,0,0,0,0};
  v8f acc10 = {0,0,0,0,0,0,0,0};
  v8f acc11 = {0,0,0,0,0,0,0,0};
  acc00 = mm16(acc00, 0, 0, 0);
  acc10 = mm16(acc10, 1, 0, 0);
  acc10 = mm16(acc10, 1, 0, 1);
  acc11 = mm16(acc11, 1, 1, 1);

  // ---- maha: y = Ti * diff, lane l computes y[l] from Ti row l in LDS ----
  float y = 0.0f;
#pragma unroll
  for (int j = 0; j < 32; ++j) {
    float dj = lane_bcast(dv, j);
    y = __builtin_fmaf(Ti[lane * LSTRIDE + j], dj, y);
  }

  // ---- reduce: trace + maha (sum of squares is layout-agnostic) ----
  float part = y * y;
#pragma unroll
  for (int i = 0; i < 8; ++i) {
    part = __builtin_fmaf(acc00[i], acc00[i], part);
    part = __builtin_fmaf(acc10[i], acc10[i], part);
    part = __builtin_fmaf(acc11[i], acc11[i], part);
  }
#pragma unroll
  for (int off = 16; off >= 1; off >>= 1)
    part += __shfl_xor(part, off, 32);

  if (lane == 0)
    out[b] = 0.5f * (part + ldq - ldp - 32.0f);
}

extern "C" void kernel_launch(void* const* d_in, const int* in_sizes, int n_in,
                              void* d_out, int out_size, void* d_ws, size_t ws_size,
                              hipStream_t stream) {
  const float* p_mu  = (const float*)d_in[0];
  const float* p_cov = (const float*)d_in[1];
  const float* q_mu  = (const float*)d_in[2];
  const float* q_cov = (const float*)d_in[3];
  float* out = (float*)d_out;
  const int nb = in_sizes[0] / 32;                 // B = 65536
  dim3 block(32 * WPB);
  dim3 grid((nb + WPB - 1) / WPB);
  kl_full_gaussian_kernel<<<grid, block, 0, stream>>>(p_mu, p_cov, q_mu, q_cov, out, nb);
}